// MLPModel_13108240187476
// MI455X (gfx1250) — compile-verified
//
#include <hip/hip_runtime.h>

// ---------------------------------------------------------------------------
// CDNA5 (gfx1250) fp32 WMMA implementation of MLP + Newton-Schulz ZCA.
// All matrix math runs on V_WMMA_F32_16X16X4_F32 (wave32).
// Inner loops use strength-reduced pointer increments (no per-iter u64 mul).
// ---------------------------------------------------------------------------

typedef float v2f __attribute__((ext_vector_type(2)));
typedef float v8f __attribute__((ext_vector_type(8)));

// 8-arg form: (neg_a, A, neg_b, B, c_mod, C, reuse_a, reuse_b)
#define WMMA_F32X4(a, b, c) \
  __builtin_amdgcn_wmma_f32_16x16x4_f32(false, (a), false, (b), (short)0, (c), false, false)

// ---------------------------------------------------------------------------
// Big GEMM: C[N,M] = op(A[N,K]) @ B[K,M]
//   SUBMEAN: subtract column-mean from A on the fly (whitening-apply pass)
//   scalePtr: optional uniform scale read from device (rsqrt(trace))
//   doRelu:   relu epilogue
// Workgroup = 256 threads (8 waves), block tile 128x128, wave tile 64x32.
// Requires N % 128 == 0, M % 128 == 0, K % 4 == 0.
// ---------------------------------------------------------------------------
template <bool SUBMEAN>
__global__ __launch_bounds__(256) void gemm_big(
    const float* __restrict__ A, const float* __restrict__ B,
    float* __restrict__ C, int N, int K, int M,
    const float* __restrict__ mean, const float* __restrict__ scalePtr,
    int doRelu) {
  const int tid  = threadIdx.x;
  const int wave = tid >> 5;
  const int lane = tid & 31;
  const int half = lane >> 4;   // 0: lanes 0-15, 1: lanes 16-31
  const int lr   = lane & 15;

  const int waveRow = wave >> 2;                    // 0..1
  const int waveCol = wave & 3;                     // 0..3
  const int rowBase = blockIdx.x * 128 + waveRow * 64;
  const int colBase = blockIdx.y * 128 + waveCol * 32;

  v8f acc[4][2];
#pragma unroll
  for (int i = 0; i < 4; ++i)
#pragma unroll
    for (int j = 0; j < 2; ++j) acc[i][j] = (v8f){};

  // Hoisted fragment base pointers; advanced by constant strides in the loop.
  const float* aPtr[4];
#pragma unroll
  for (int i = 0; i < 4; ++i)
    aPtr[i] = A + (size_t)(rowBase + i * 16 + lr) * K + half * 2;
  const float* bPtr[2];
#pragma unroll
  for (int j = 0; j < 2; ++j)
    bPtr[j] = B + (size_t)(half * 2) * M + (colBase + j * 16 + lr);
  const float* mPtr = SUBMEAN ? (mean + half * 2) : nullptr;
  const size_t bStep = (size_t)4 * M;

  for (int k = 0; k < K; k += 4) {
    v2f a[4];
#pragma unroll
    for (int i = 0; i < 4; ++i) {
      v2f av = *(const v2f*)aPtr[i];      // 8B-aligned (K even, half*2 even)
      aPtr[i] += 4;
      if (SUBMEAN) { av.x -= mPtr[0]; av.y -= mPtr[1]; }
      a[i] = av;
    }
    if (SUBMEAN) mPtr += 4;
    v2f b[2];
#pragma unroll
    for (int j = 0; j < 2; ++j) {
      v2f bv;
      bv.x = bPtr[j][0];
      bv.y = bPtr[j][M];
      b[j] = bv;
      bPtr[j] += bStep;
    }
#pragma unroll
    for (int i = 0; i < 4; ++i)
#pragma unroll
      for (int j = 0; j < 2; ++j) acc[i][j] = WMMA_F32X4(a[i], b[j], acc[i][j]);
  }

  const float s = scalePtr ? scalePtr[0] : 1.0f;
#pragma unroll
  for (int i = 0; i < 4; ++i)
#pragma unroll
    for (int j = 0; j < 2; ++j)
#pragma unroll
      for (int v = 0; v < 8; ++v) {
        // C/D layout: VGPR v holds row (v + half*8) of the 16x16 tile
        const int row = rowBase + i * 16 + half * 8 + v;
        const int col = colBase + j * 16 + lr;
        float val = acc[i][j][v] * s;
        if (doRelu) val = fmaxf(val, 0.0f);
        C[(size_t)row * M + col] = val;
      }
}

// ---------------------------------------------------------------------------
// Gram matrix (split-K): G[d,d] += H[chunk,d]^T @ H[chunk,d]
// blockIdx.z selects the K-chunk over N; partials merged with f32 atomics.
// Requires d == 128 or d % 128 == 0; chunk % 4 == 0.
// ---------------------------------------------------------------------------
__global__ __launch_bounds__(256) void cov_gemm(
    const float* __restrict__ H, float* __restrict__ G, int d, int chunk) {
  const int tid  = threadIdx.x;
  const int wave = tid >> 5;
  const int lane = tid & 31;
  const int half = lane >> 4;
  const int lr   = lane & 15;

  const int waveRow = wave >> 2;
  const int waveCol = wave & 3;
  const int rowBase = blockIdx.x * 128 + waveRow * 64;
  const int colBase = blockIdx.y * 128 + waveCol * 32;
  const int kbase   = blockIdx.z * chunk;

  v8f acc[4][2];
#pragma unroll
  for (int i = 0; i < 4; ++i)
#pragma unroll
    for (int j = 0; j < 2; ++j) acc[i][j] = (v8f){};

  // Row pointer for this half-wave's K pair; advanced by 4 rows per step.
  const float* r0 = H + (size_t)(kbase + half * 2) * d;
  const size_t rStep = (size_t)4 * d;
  int m0 = rowBase + lr;
  int n0 = colBase + lr;

  for (int k = 0; k < chunk; k += 4) {
    const float* r1 = r0 + d;
    v2f a[4], b[2];
#pragma unroll
    for (int i = 0; i < 4; ++i) {
      a[i].x = r0[m0 + i * 16];
      a[i].y = r1[m0 + i * 16];
    }
#pragma unroll
    for (int j = 0; j < 2; ++j) {
      b[j].x = r0[n0 + j * 16];
      b[j].y = r1[n0 + j * 16];
    }
    r0 += rStep;
#pragma unroll
    for (int i = 0; i < 4; ++i)
#pragma unroll
      for (int j = 0; j < 2; ++j) acc[i][j] = WMMA_F32X4(a[i], b[j], acc[i][j]);
  }

#pragma unroll
  for (int i = 0; i < 4; ++i)
#pragma unroll
    for (int j = 0; j < 2; ++j)
#pragma unroll
      for (int v = 0; v < 8; ++v) {
        const int row = rowBase + i * 16 + half * 8 + v;
        const int col = colBase + j * 16 + lr;
        atomicAdd(&G[(size_t)row * d + col], acc[i][j][v]);
      }
}

// ---------------------------------------------------------------------------
// Small d x d GEMM (d in {128,256}): D = alpha*A@B + beta*Cin
// One wave per 16x16 output tile. Used for the Newton-Schulz iterations.
// ---------------------------------------------------------------------------
__global__ __launch_bounds__(32) void small_gemm(
    const float* __restrict__ A, const float* __restrict__ B,
    const float* __restrict__ Cin, float* __restrict__ D, int d,
    float alpha, float beta) {
  const int lane = threadIdx.x & 31;
  const int half = lane >> 4;
  const int lr   = lane & 15;
  const int row0 = blockIdx.x * 16;
  const int col0 = blockIdx.y * 16;

  v8f acc = (v8f){};
  const float* aPtr = A + (size_t)(row0 + lr) * d + half * 2;
  const float* bPtr = B + (size_t)(half * 2) * d + col0 + lr;
  const size_t bStep = (size_t)4 * d;

  for (int k = 0; k < d; k += 4) {
    v2f a = *(const v2f*)aPtr;
    aPtr += 4;
    v2f b;
    b.x = bPtr[0];
    b.y = bPtr[d];
    bPtr += bStep;
    acc = WMMA_F32X4(a, b, acc);
  }
#pragma unroll
  for (int v = 0; v < 8; ++v) {
    const int row = row0 + half * 8 + v;
    const int col = col0 + lr;
    float val = alpha * acc[v];
    if (Cin) val += beta * Cin[(size_t)row * d + col];
    D[(size_t)row * d + col] = val;
  }
}

// ---------------------------------------------------------------------------
// Utility kernels
// ---------------------------------------------------------------------------
__global__ void zero_kernel(float* __restrict__ p, size_t n) {
  for (size_t i = (size_t)blockIdx.x * blockDim.x + threadIdx.x; i < n;
       i += (size_t)gridDim.x * blockDim.x)
    p[i] = 0.0f;
}

// blockDim.x == d ; each block sums `rowsPerBlock` rows, one atomic per column
__global__ void col_mean(const float* __restrict__ H, float* __restrict__ acc,
                         int rowsPerBlock, int N, int d) {
  const int col = threadIdx.x;
  const int r0  = blockIdx.x * rowsPerBlock;
  int r1 = r0 + rowsPerBlock;
  if (r1 > N) r1 = N;
  float s = 0.0f;
  const float* p = H + (size_t)r0 * d + col;
  for (int r = r0; r < r1; ++r) {
    s += *p;
    p += d;
  }
  atomicAdd(&acc[col], s);
}

__global__ void vec_scale(float* __restrict__ v, float s, int n) {
  const int i = blockIdx.x * blockDim.x + threadIdx.x;
  if (i < n) v[i] *= s;
}

// cov[j,k] = G[j,k]/N - m[j]*m[k] + (j==k)*0.1   (grid = d blocks of d threads)
__global__ void finalize_cov(const float* __restrict__ G,
                             const float* __restrict__ m,
                             float* __restrict__ cov, int d, float invN) {
  const int j = blockIdx.x;
  const int k = threadIdx.x;
  float v = G[(size_t)j * d + k] * invN - m[j] * m[k];
  if (j == k) v += 0.1f;  // eye(d) / ZCA_TEMP
  cov[(size_t)j * d + k] = v;
}

// One block of 256 threads: tr = sum(diag(cov)); cov /= tr; s = rsqrt(tr)
__global__ __launch_bounds__(256) void trace_norm(float* __restrict__ cov,
                                                  float* __restrict__ trOut,
                                                  float* __restrict__ sOut,
                                                  int d) {
  __shared__ float red[256];
  const int t = threadIdx.x;
  red[t] = (t < d) ? cov[(size_t)t * d + t] : 0.0f;
  __syncthreads();
  for (int off = 128; off > 0; off >>= 1) {
    if (t < off) red[t] += red[t + off];
    __syncthreads();
  }
  if (t == 0) {
    trOut[0] = red[0];
    sOut[0]  = rsqrtf(red[0]);
  }
  __syncthreads();
  const float inv = 1.0f / red[0];
  const int total = d * d;
  for (int i = t; i < total; i += 256) cov[i] *= inv;
}

__global__ void set_identity(float* __restrict__ P, int d) {
  const int j = blockIdx.x;
  const int k = threadIdx.x;
  P[(size_t)j * d + k] = (j == k) ? 1.0f : 0.0f;
}

// ---------------------------------------------------------------------------
// Host orchestration
// ---------------------------------------------------------------------------
extern "C" void kernel_launch(void* const* d_in, const int* in_sizes, int n_in,
                              void* d_out, int out_size, void* d_ws,
                              size_t ws_size, hipStream_t stream) {
  (void)n_in; (void)out_size; (void)ws_size;
  const float* x  = (const float*)d_in[0];
  const float* W0 = (const float*)d_in[1];
  const float* W1 = (const float*)d_in[2];
  const float* W2 = (const float*)d_in[3];

  const int Din  = 512;
  const int Dh   = 256;
  const int Dout = 128;
  const int N    = in_sizes[0] / Din;  // 400000 (multiple of 128)

  // Workspace layout (floats)
  float* ws = (float*)d_ws;
  size_t off = 0;
  float* bufA = ws + off; off += (size_t)N * Dh;   // GEMM outputs (pre-ZCA)
  float* bufB = ws + off; off += (size_t)N * Dh;   // ZCA outputs (post-relu)
  float* mean = ws + off; off += Dh;
  float* G    = ws + off; off += (size_t)Dh * Dh;
  float* Cov  = ws + off; off += (size_t)Dh * Dh;  // becomes cov_n in place
  float* P    = ws + off; off += (size_t)Dh * Dh;
  float* T1   = ws + off; off += (size_t)Dh * Dh;
  float* T2   = ws + off; off += (size_t)Dh * Dh;
  float* trB  = ws + off; off += 1;
  float* sB   = ws + off; off += 1;

  const float invN = 1.0f / (float)N;

  auto run_zca = [&](const float* h, float* out, int d) {
    zero_kernel<<<8, 256, 0, stream>>>(mean, (size_t)d);
    zero_kernel<<<256, 256, 0, stream>>>(G, (size_t)d * d);

    const int rowsPerBlock = 400;
    col_mean<<<(N + rowsPerBlock - 1) / rowsPerBlock, d, 0, stream>>>(
        h, mean, rowsPerBlock, N, d);
    vec_scale<<<1, d, 0, stream>>>(mean, invN, d);

    const int splits = 125;             // N/125 = 3200, multiple of 4
    const int chunk  = N / splits;
    dim3 cgrid(d / 128, d / 128, splits);
    cov_gemm<<<cgrid, 256, 0, stream>>>(h, G, d, chunk);

    finalize_cov<<<d, d, 0, stream>>>(G, mean, Cov, d, invN);
    trace_norm<<<1, 256, 0, stream>>>(Cov, trB, sB, d);
    set_identity<<<d, d, 0, stream>>>(P, d);

    dim3 sgrid(d / 16, d / 16);
    for (int it = 0; it < 10; ++it) {
      // T1 = P@P ; T2 = T1@P = P^3 ; P = 1.5*P - 0.5*(T2 @ cov_n)
      small_gemm<<<sgrid, 32, 0, stream>>>(P, P, nullptr, T1, d, 1.0f, 0.0f);
      small_gemm<<<sgrid, 32, 0, stream>>>(T1, P, nullptr, T2, d, 1.0f, 0.0f);
      small_gemm<<<sgrid, 32, 0, stream>>>(T2, Cov, P, P, d, -0.5f, 1.5f);
    }

    // out = relu( rsqrt(tr) * (h - mean) @ P )
    dim3 agrid(N / 128, d / 128);
    gemm_big<true><<<agrid, 256, 0, stream>>>(h, P, out, N, d, d, mean, sB, 1);
  };

  // Layer 0: h = x @ W0 ; ZCA(256) ; relu
  gemm_big<false><<<dim3(N / 128, Dh / 128), 256, 0, stream>>>(
      x, W0, bufA, N, Din, Dh, nullptr, nullptr, 0);
  run_zca(bufA, bufB, Dh);

  // Layer 1: h = bufB @ W1 ; ZCA(256) ; relu
  gemm_big<false><<<dim3(N / 128, Dh / 128), 256, 0, stream>>>(
      bufB, W1, bufA, N, Dh, Dh, nullptr, nullptr, 0);
  run_zca(bufA, bufB, Dh);

  // Layer 2: h = bufB @ W2 ; ZCA(128) ; relu -> d_out
  gemm_big<false><<<dim3(N / 128, Dout / 128), 256, 0, stream>>>(
      bufB, W2, bufA, N, Dh, Dout, nullptr, nullptr, 0);
  run_zca(bufA, (float*)d_out, Dout);
}